// Gate_33827162423867
// MI455X (gfx1250) — compile-verified
//
#include <hip/hip_runtime.h>
#include <math.h>
#include <stdint.h>

// ---- problem constants (from reference) ----
#define TOKENS 16384
#define DIM    4096
#define NEXP   64
#define TOPK   8

// ---- GEMM tiling ----
#define KC        64                    // K-chunk staged in LDS (double buffered)
#define LSTRIDE   (KC + 4)              // pad: 16B-aligned rows + bank spread
#define BUF_FLTS  (NEXP * LSTRIDE)      // 4352 floats = 17408 B per buffer
#define NCHUNK    (DIM / KC)            // 64

typedef float v2f __attribute__((ext_vector_type(2)));
typedef float v4f __attribute__((ext_vector_type(4)));
typedef float v8f __attribute__((ext_vector_type(8)));

// CDNA5 async DMA: global -> LDS, 16B per lane, tracked by ASYNCcnt.
__device__ __forceinline__ void async_ld_b128(uint32_t lds_byte, const float* gaddr) {
    asm volatile("global_load_async_to_lds_b128 %0, %1, off"
                 :: "v"(lds_byte), "v"(gaddr)
                 : "memory");
}

// Stage weights[0:64][kc:kc+KC] into LDS buffer at buf_byte.
// 32 async b128 per block; each wave issues 4 (ASYNCcnt += 4).
// Instruction j covers expert rows 2*(wave*4+j) and 2*(wave*4+j)+1:
//   lanes 0-15 -> even row, lanes 16-31 -> odd row, 16B per lane.
__device__ __forceinline__ void stage_chunk(const float* __restrict__ w, int kc,
                                            uint32_t buf_byte, int wave, int lo, int hi) {
    #pragma unroll
    for (int j = 0; j < 4; ++j) {
        const int e = 2 * (wave * 4 + j) + hi;
        const uint32_t lds = buf_byte + (uint32_t)(e * LSTRIDE + lo * 4) * 4u;
        const float* g = w + (size_t)e * DIM + kc + lo * 4;
        async_ld_b128(lds, g);
    }
}

// scores[t][e] = sum_d x[t][d] * w[e][d]
// block = 256 threads = 8 waves; wave -> 16 tokens x 64 experts.
__global__ __launch_bounds__(256)
void gate_gemm_kernel(const float* __restrict__ x,
                      const float* __restrict__ w,
                      float* __restrict__ scores) {
    __shared__ float lds[2 * BUF_FLTS];
    const uint32_t lds_base = (uint32_t)(uintptr_t)(void*)&lds[0]; // AS3 byte offset

    const int tid   = threadIdx.x;
    const int wave  = tid >> 5;
    const int lane  = tid & 31;
    const int lo    = lane & 15;     // M (A) / N (B,C)
    const int hi    = lane >> 4;     // K-pair select (A/B), M+8 (C/D)
    const int khalf = hi << 1;

    const int tokBase = blockIdx.x * 128 + wave * 16;
    const float* arow = x + (size_t)(tokBase + lo) * DIM + khalf;

    v8f acc0 = {}, acc1 = {}, acc2 = {}, acc3 = {};

    // prologue: kick chunk 0 into buffer 0
    stage_chunk(w, 0, lds_base, wave, lo, hi);

    for (int c = 0; c < NCHUNK; ++c) {
        // WAR: everyone finished computing on buf (c+1)&1 (iteration c-1)
        __syncthreads();
        if (c + 1 < NCHUNK) {
            stage_chunk(w, (c + 1) * KC,
                        lds_base + (uint32_t)(((c + 1) & 1) * BUF_FLTS * 4),
                        wave, lo, hi);
            // allow next chunk's 4 ops outstanding; chunk c's (older, in-order) are done
            asm volatile("s_wait_asynccnt 0x4" ::: "memory");
        } else {
            asm volatile("s_wait_asynccnt 0x0" ::: "memory");
        }
        // all waves' staging quotas for chunk c complete
        __syncthreads();

        const float* buf = &lds[(c & 1) * BUF_FLTS];
        const float* ak  = arow + c * KC;

        #pragma unroll 4
        for (int k = 0; k < KC; k += 4) {
            // A fragment: streamed once from HBM -> non-temporal (don't pollute L2)
            v2f a = __builtin_nontemporal_load((const v2f*)(ak + k));
            // B fragments from LDS: lane lo = expert column N, khalf selects K pair
            v2f b0 = *(const v2f*)&buf[(0 * 16 + lo) * LSTRIDE + k + khalf];
            v2f b1 = *(const v2f*)&buf[(1 * 16 + lo) * LSTRIDE + k + khalf];
            v2f b2 = *(const v2f*)&buf[(2 * 16 + lo) * LSTRIDE + k + khalf];
            v2f b3 = *(const v2f*)&buf[(3 * 16 + lo) * LSTRIDE + k + khalf];

            acc0 = __builtin_amdgcn_wmma_f32_16x16x4_f32(false, a, false, b0,
                                                         (short)0, acc0, false, false);
            acc1 = __builtin_amdgcn_wmma_f32_16x16x4_f32(false, a, false, b1,
                                                         (short)0, acc1, false, false);
            acc2 = __builtin_amdgcn_wmma_f32_16x16x4_f32(false, a, false, b2,
                                                         (short)0, acc2, false, false);
            acc3 = __builtin_amdgcn_wmma_f32_16x16x4_f32(false, a, false, b3,
                                                         (short)0, acc3, false, false);
        }
    }

    // D layout: VGPR v -> M = v + 8*hi, lane lo -> N
    float* outBase = scores + (size_t)tokBase * NEXP;
    #pragma unroll
    for (int v = 0; v < 8; ++v) {
        const int m = v + 8 * hi;
        float* row = outBase + (size_t)m * NEXP;
        row[0 * 16 + lo] = acc0[v];
        row[1 * 16 + lo] = acc1[v];
        row[2 * 16 + lo] = acc2[v];
        row[3 * 16 + lo] = acc3[v];
    }
}

// one thread per token: softmax over 64 experts + stable top-8
__global__ __launch_bounds__(256)
void gate_topk_kernel(const float* __restrict__ scores,
                      float* __restrict__ gate_out,
                      int* __restrict__ idx_out) {
    const int t = blockIdx.x * blockDim.x + threadIdx.x;
    if (t >= TOKENS) return;

    const float* s = scores + (size_t)t * NEXP;
    float v[NEXP];
    #pragma unroll
    for (int i = 0; i < NEXP; i += 4) {
        v4f q = *(const v4f*)(s + i);
        v[i + 0] = q.x; v[i + 1] = q.y; v[i + 2] = q.z; v[i + 3] = q.w;
    }

    float m = v[0];
    #pragma unroll
    for (int i = 1; i < NEXP; ++i) m = fmaxf(m, v[i]);

    float denom = 0.0f;
    #pragma unroll
    for (int i = 0; i < NEXP; ++i) denom += __expf(v[i] - m);
    const float inv_denom = 1.0f / denom;

    float bv[TOPK];
    int   bi[TOPK];
    #pragma unroll
    for (int j = 0; j < TOPK; ++j) { bv[j] = -INFINITY; bi[j] = 0; }

    // strict '>' insertion keeps earliest index first on ties (matches lax.top_k)
    #pragma unroll
    for (int i = 0; i < NEXP; ++i) {
        float val = v[i];
        if (val > bv[TOPK - 1]) {
            bv[TOPK - 1] = val;
            bi[TOPK - 1] = i;
            #pragma unroll
            for (int j = TOPK - 1; j > 0; --j) {
                if (bv[j] > bv[j - 1]) {
                    float tv = bv[j]; bv[j] = bv[j - 1]; bv[j - 1] = tv;
                    int   ti = bi[j]; bi[j] = bi[j - 1]; bi[j - 1] = ti;
                }
            }
        }
    }

    #pragma unroll
    for (int j = 0; j < TOPK; ++j) {
        gate_out[(size_t)t * TOPK + j] = __expf(bv[j] - m) * inv_denom; // ROUTE_SCALE=1
        idx_out[(size_t)t * TOPK + j]  = bi[j];
    }
}

extern "C" void kernel_launch(void* const* d_in, const int* in_sizes, int n_in,
                              void* d_out, int out_size, void* d_ws, size_t ws_size,
                              hipStream_t stream) {
    const float* x = (const float*)d_in[0];
    const float* w = (const float*)d_in[1];
    // d_in[2] = bias: dead code in reference, unused.

    float* scores = (float*)d_ws;                                   // 16384*64 f32 = 4 MB
    float* gate   = (float*)d_out;                                  // [16384, 8] f32
    int*   idx    = (int*)((float*)d_out + (size_t)TOKENS * TOPK);  // [16384, 8] i32

    gate_gemm_kernel<<<TOKENS / 128, 256, 0, stream>>>(x, w, scores);
    gate_topk_kernel<<<TOKENS / 256, 256, 0, stream>>>(scores, gate, idx);
}